// HeteroGNN_89412629168563
// MI455X (gfx1250) — compile-verified
//
#include <hip/hip_runtime.h>
#include <hip/hip_bf16.h>

#define N_USER 100000
#define N_GAME 20000
#define N_EDGE 1600000
#define DU 64
#define DG 128
#define H  128

typedef __attribute__((ext_vector_type(2))) float v2f;
typedef __attribute__((ext_vector_type(8))) float v8f;

// ---------------------------------------------------------------------------
// WMMA helper: accumulate C += X[mBase:mBase+16, :] @ W[:, nBase:nBase+16]
// using V_WMMA_F32_16X16X4_F32 (exact fp32). X is [rows, K] row-major,
// W is [K, 128] row-major.
//
// Per-lane fragment layout (CDNA5 ISA 7.12.2, 32-bit 16x4 A-matrix):
//   lanes 0-15 : M = lane,    v0 = K+0, v1 = K+1
//   lanes 16-31: M = lane-16, v0 = K+2, v1 = K+3
// B (4x16) mirrored with N = lane&15.
// ---------------------------------------------------------------------------
template <int K>
__device__ __forceinline__ v8f wmma_acc_f32(const float* __restrict__ X,
                                            const float* __restrict__ W,
                                            int mBase, int nBase, int lane,
                                            v8f c) {
    const int mr   = mBase + (lane & 15);
    const int n    = nBase + (lane & 15);
    const int kSel = (lane >> 4) << 1;  // 0 for lanes 0-15, 2 for lanes 16-31
    const float* xrow = X + (size_t)mr * K;
#pragma unroll
    for (int k = 0; k < K; k += 4) {
        const int kk = k + kSel;
        v2f a, b;
        a.x = xrow[kk];
        a.y = xrow[kk + 1];
        b.x = W[(size_t)kk * H + n];
        b.y = W[(size_t)(kk + 1) * H + n];
        c = __builtin_amdgcn_wmma_f32_16x16x4_f32(
                /*neg_a=*/false, a, /*neg_b=*/false, b,
                /*c_mod=*/(short)0, c, /*reuse_a=*/false, /*reuse_b=*/false);
    }
    return c;
}

// out[mBase+16, 128] = act(X @ W + bias). Block = 256 threads = 8 waves,
// wave w computes N-tile [16w, 16w+16). Grid.x = rows/16.
template <int K, bool RELU>
__global__ void gemm_bias_act_kernel(const float* __restrict__ X,
                                     const float* __restrict__ W,
                                     const float* __restrict__ bias,
                                     float* __restrict__ out) {
    const int lane  = threadIdx.x & 31;
    const int wave  = threadIdx.x >> 5;
    const int mBase = blockIdx.x * 16;
    const int nBase = wave * 16;
    const int n     = nBase + (lane & 15);

    v8f c;
    const float bv = bias[n];
#pragma unroll
    for (int i = 0; i < 8; ++i) c[i] = bv;

    c = wmma_acc_f32<K>(X, W, mBase, nBase, lane, c);

    // C/D layout: VGPR i -> M = mBase + i (+8 for upper lane half), N = lane&15
    const int rBase = mBase + ((lane >> 4) << 3);
#pragma unroll
    for (int i = 0; i < 8; ++i) {
        float v = c[i];
        if (RELU) v = fmaxf(v, 0.0f);
        out[(size_t)(rBase + i) * H + n] = v;
    }
}

// out = M @ Wl + bias + Hf @ Wr   (both K = 128, no activation)
__global__ void dual_gemm_bias_kernel(const float* __restrict__ M,
                                      const float* __restrict__ Wl,
                                      const float* __restrict__ bias,
                                      const float* __restrict__ Hf,
                                      const float* __restrict__ Wr,
                                      float* __restrict__ out) {
    const int lane  = threadIdx.x & 31;
    const int wave  = threadIdx.x >> 5;
    const int mBase = blockIdx.x * 16;
    const int nBase = wave * 16;
    const int n     = nBase + (lane & 15);

    v8f c;
    const float bv = bias[n];
#pragma unroll
    for (int i = 0; i < 8; ++i) c[i] = bv;

    c = wmma_acc_f32<H>(M,  Wl, mBase, nBase, lane, c);
    c = wmma_acc_f32<H>(Hf, Wr, mBase, nBase, lane, c);

    const int rBase = mBase + ((lane >> 4) << 3);
#pragma unroll
    for (int i = 0; i < 8; ++i)
        out[(size_t)(rBase + i) * H + n] = c[i];
}

// ---------------------------------------------------------------------------
// Edge scatter: one wave per edge, 4 floats per lane (128 total).
// acc[dst] += feat[src]; cnt[dst] += 1.  192MB L2 keeps this on-die.
// ---------------------------------------------------------------------------
__global__ void scatter_accum_kernel(const float* __restrict__ feat,
                                     const int* __restrict__ src,
                                     const int* __restrict__ dst,
                                     float* __restrict__ acc,
                                     float* __restrict__ cnt,
                                     int nEdges) {
    const int e = blockIdx.x * (blockDim.x >> 5) + (threadIdx.x >> 5);
    if (e >= nEdges) return;
    const int lane = threadIdx.x & 31;
    const int s = src[e];
    const int d = dst[e];

    const float4 v = ((const float4*)(feat + (size_t)s * H))[lane];
    float* ap = acc + (size_t)d * H + lane * 4;
    unsafeAtomicAdd(ap + 0, v.x);
    unsafeAtomicAdd(ap + 1, v.y);
    unsafeAtomicAdd(ap + 2, v.z);
    unsafeAtomicAdd(ap + 3, v.w);
    if (lane == 0) unsafeAtomicAdd(cnt + d, 1.0f);
}

// acc[i,:] /= max(cnt[i], 1)
__global__ void seg_div_kernel(float* __restrict__ acc,
                               const float* __restrict__ cnt,
                               long total) {
    long i = (long)blockIdx.x * blockDim.x + threadIdx.x;
    if (i >= total) return;
    acc[i] = acc[i] / fmaxf(cnt[i >> 7], 1.0f);
}

extern "C" void kernel_launch(void* const* d_in, const int* in_sizes, int n_in,
                              void* d_out, int out_size, void* d_ws, size_t ws_size,
                              hipStream_t stream) {
    const float* x_user = (const float*)d_in[0];
    const float* x_game = (const float*)d_in[1];
    const float* W_user = (const float*)d_in[2];
    const float* b_user = (const float*)d_in[3];
    const float* W_game = (const float*)d_in[4];
    const float* b_game = (const float*)d_in[5];
    const float* Wl_ug  = (const float*)d_in[6];
    const float* bl_ug  = (const float*)d_in[7];
    const float* Wr_ug  = (const float*)d_in[8];
    const float* Wl_gu  = (const float*)d_in[9];
    const float* bl_gu  = (const float*)d_in[10];
    const float* Wr_gu  = (const float*)d_in[11];
    const int*   src_ug = (const int*)d_in[12];
    const int*   dst_ug = (const int*)d_in[13];
    const int*   src_gu = (const int*)d_in[14];
    const int*   dst_gu = (const int*)d_in[15];

    float* out_user = (float*)d_out;                       // [N_USER, H]
    float* out_game = out_user + (size_t)N_USER * H;       // [N_GAME, H]

    // Workspace layout (floats)
    float* hu    = (float*)d_ws;                 // [N_USER, H]
    float* hg    = hu    + (size_t)N_USER * H;   // [N_GAME, H]
    float* acc_g = hg    + (size_t)N_GAME * H;   // [N_GAME, H]
    float* cnt_g = acc_g + (size_t)N_GAME * H;   // [N_GAME]
    float* acc_u = cnt_g + N_GAME;               // [N_USER, H]
    float* cnt_u = acc_u + (size_t)N_USER * H;   // [N_USER]

    // Zero the accumulators + counts (contiguous region), graph-capture safe.
    hipMemsetAsync(acc_g, 0,
                   sizeof(float) * ((size_t)N_GAME * H + N_GAME +
                                    (size_t)N_USER * H + N_USER),
                   stream);

    // Input projections + ReLU (WMMA f32)
    gemm_bias_act_kernel<DU, true><<<N_USER / 16, 256, 0, stream>>>(
        x_user, W_user, b_user, hu);
    gemm_bias_act_kernel<DG, true><<<N_GAME / 16, 256, 0, stream>>>(
        x_game, W_game, b_game, hg);

    // Edge mean-aggregation (atomic accumulate, then divide)
    const int edgeBlocks = (N_EDGE + 7) / 8;   // 8 waves (edges) per block
    scatter_accum_kernel<<<edgeBlocks, 256, 0, stream>>>(
        hu, src_ug, dst_ug, acc_g, cnt_g, N_EDGE);
    scatter_accum_kernel<<<edgeBlocks, 256, 0, stream>>>(
        hg, src_gu, dst_gu, acc_u, cnt_u, N_EDGE);

    const long ng_tot = (long)N_GAME * H;
    const long nu_tot = (long)N_USER * H;
    seg_div_kernel<<<(int)((ng_tot + 255) / 256), 256, 0, stream>>>(acc_g, cnt_g, ng_tot);
    seg_div_kernel<<<(int)((nu_tot + 255) / 256), 256, 0, stream>>>(acc_u, cnt_u, nu_tot);

    // SAGE output layers (dual WMMA GEMM + bias)
    dual_gemm_bias_kernel<<<N_GAME / 16, 256, 0, stream>>>(
        acc_g, Wl_ug, bl_ug, hg, Wr_ug, out_game);
    dual_gemm_bias_kernel<<<N_USER / 16, 256, 0, stream>>>(
        acc_u, Wl_gu, bl_gu, hu, Wr_gu, out_user);
}